// GraphConv_60971355734091
// MI455X (gfx1250) — compile-verified
//
#include <hip/hip_runtime.h>
#include <math.h>

// ---------------------------------------------------------------------------
// GraphConv attention layer for MI455X (gfx1250), wave32 + WMMA f32 16x16x4.
//
//   H    = x @ W^T                                   (WMMA f32 GEMM)
//   e    = combined double-normalization scale       (two graph passes)
//   c1/c2= leaky_relu(H @ t_s), leaky_relu(H @ t_r)
//   m,l  = rowwise softmax stats of masked scores    (one graph pass)
//   out  = 0.5*(softmax(S) @ H + H)                  (fused: A fragment built
//           in registers per WMMA k-step; A is never materialized in memory)
//
// softmax rows sum to 1 -> (A+I) rowsum == 2 -> final norm is just *0.5.
// HBM traffic ~256 MB total -> ~11 us floor at 23.3 TB/s.
// ---------------------------------------------------------------------------

typedef float v2f __attribute__((ext_vector_type(2)));
typedef float v8f __attribute__((ext_vector_type(8)));

#define GC_N     4096
#define GC_INF   512
#define GC_OUTF  256
#define GC_NEG   (-1e11f)

static __device__ __forceinline__ v8f wmma_f32(v2f a, v2f b, v8f c) {
  // V_WMMA_F32_16X16X4_F32 : D = A(16x4) * B(4x16) + C(16x16), fp32 throughout
  return __builtin_amdgcn_wmma_f32_16x16x4_f32(
      /*neg_a=*/false, a, /*neg_b=*/false, b,
      /*c_mod=*/(short)0, c, /*reuse_a=*/false, /*reuse_b=*/false);
}

// ---------------------------------------------------------------------------
// Kernel 1: H[N, OUTF] = x[N, INF] @ W[OUTF, INF]^T
// One wave per 16x16 output tile. 4096 tiles -> 1024 blocks x 128 threads.
// ---------------------------------------------------------------------------
__global__ void gc_gemm_xwT(const float* __restrict__ x,
                            const float* __restrict__ w,
                            float* __restrict__ H) {
  const int wave = blockIdx.x * 4 + (threadIdx.x >> 5);
  const int lane = threadIdx.x & 31;
  const int tm = wave >> 4;            // 0..255  (M tiles)
  const int tn = wave & 15;            // 0..15   (N tiles)
  const int l16 = lane & 15;
  const int khalf = (lane >> 4) << 1;  // 0 or 2

  const int m = (tm << 4) + l16;
  const int n = (tn << 4) + l16;

  const float* __restrict__ arow = x + (size_t)m * GC_INF + khalf;
  const float* __restrict__ brow = w + (size_t)n * GC_INF + khalf;  // B[k][n] = W[n][k]

  v8f acc = {};
#pragma unroll 8
  for (int k = 0; k < GC_INF; k += 4) {
    v2f a = *(const v2f*)(arow + k);
    v2f b = *(const v2f*)(brow + k);
    acc = wmma_f32(a, b, acc);
  }

  const int row0 = (tm << 4) + ((lane >> 4) << 3);  // lanes 16..31 hold M+8
#pragma unroll
  for (int r = 0; r < 8; ++r)
    H[(size_t)(row0 + r) * GC_OUTF + n] = acc[r];
}

// ---------------------------------------------------------------------------
// Kernel 2: c1[i] = leaky(H[i,:] . t_s), c2[i] = leaky(H[i,:] . t_r)
// One block (256 threads) per row.
// ---------------------------------------------------------------------------
__global__ void gc_rowdot(const float* __restrict__ H,
                          const float* __restrict__ ts,
                          const float* __restrict__ tr,
                          float* __restrict__ c1,
                          float* __restrict__ c2) {
  __shared__ float r1[256];
  __shared__ float r2[256];
  const int i = blockIdx.x;
  const int t = threadIdx.x;
  const float h = H[(size_t)i * GC_OUTF + t];
  r1[t] = h * ts[t];
  r2[t] = h * tr[t];
  __syncthreads();
  for (int off = 128; off > 0; off >>= 1) {
    if (t < off) { r1[t] += r1[t + off]; r2[t] += r2[t + off]; }
    __syncthreads();
  }
  if (t == 0) {
    float a = r1[0], b = r2[0];
    c1[i] = a > 0.f ? a : 0.2f * a;   // leaky_relu slope 0.2
    c2[i] = b > 0.f ? b : 0.2f * b;
  }
}

// ---------------------------------------------------------------------------
// Kernel 3: d1[i] = 1/sqrt(fix(rowsum(g - diag)))
// ---------------------------------------------------------------------------
__global__ void gc_deg1(const float* __restrict__ g, float* __restrict__ d1) {
  __shared__ float red[256];
  const int i = blockIdx.x;
  const int t = threadIdx.x;
  const float* __restrict__ row = g + (size_t)i * GC_N;
  float s = 0.f;
#pragma unroll
  for (int u = 0; u < GC_N / 256; ++u) {
    int j = t + (u << 8);
    s += (j == i) ? 0.f : row[j];
  }
  red[t] = s;
  __syncthreads();
  for (int off = 128; off > 0; off >>= 1) {
    if (t < off) red[t] += red[t + off];
    __syncthreads();
  }
  if (t == 0) {
    float ss = red[0];
    if (ss == 0.f) ss = 1.f;
    d1[i] = 1.0f / sqrtf(ss);
  }
}

// ---------------------------------------------------------------------------
// Kernel 4: s2[i] = d1[i] * sum_j g'[i,j]*d1[j];  e[i] = d1[i]/sqrt(fix(s2))
// ---------------------------------------------------------------------------
__global__ void gc_deg2(const float* __restrict__ g,
                        const float* __restrict__ d1,
                        float* __restrict__ e) {
  __shared__ float red[256];
  const int i = blockIdx.x;
  const int t = threadIdx.x;
  const float* __restrict__ row = g + (size_t)i * GC_N;
  float s = 0.f;
#pragma unroll
  for (int u = 0; u < GC_N / 256; ++u) {
    int j = t + (u << 8);
    s += (j == i) ? 0.f : row[j] * d1[j];
  }
  red[t] = s;
  __syncthreads();
  for (int off = 128; off > 0; off >>= 1) {
    if (t < off) red[t] += red[t + off];
    __syncthreads();
  }
  if (t == 0) {
    float di = d1[i];
    float s2 = di * red[0];
    if (s2 == 0.f) s2 = 1.f;
    e[i] = di / sqrtf(s2);        // e = d1 * d2
  }
}

// ---------------------------------------------------------------------------
// Kernel 5: per-row softmax stats over masked scores:
//   S[i,j] = g'_ij > 0 ? e_i*e_j*g'_ij*(c1_i+c2_j)*0.5 : NEG_FILL
//   sm[i] = max_j S[i,j];  sinvl[i] = 1 / sum_j exp(S[i,j]-sm[i])
// One block (256 threads) per row; 16 scores/thread held in registers.
// ---------------------------------------------------------------------------
__global__ void gc_attn_stats(const float* __restrict__ g,
                              const float* __restrict__ e,
                              const float* __restrict__ c1,
                              const float* __restrict__ c2,
                              float* __restrict__ sm,
                              float* __restrict__ sinvl) {
  __shared__ float red[256];
  const int i = blockIdx.x;
  const int t = threadIdx.x;
  const float* __restrict__ grow = g + (size_t)i * GC_N;
  const float ei = e[i];
  const float ci = c1[i];

  float sv[GC_N / 256];
  float m = -3.4e38f;
#pragma unroll
  for (int u = 0; u < GC_N / 256; ++u) {
    int j = t + (u << 8);
    float gij = (j == i) ? 0.f : grow[j];   // diagonal removed
    float s = (gij > 0.f)
                  ? (ei * e[j] * gij) * (ci + c2[j]) * 0.5f
                  : GC_NEG;
    sv[u] = s;
    m = fmaxf(m, s);
  }
  red[t] = m;
  __syncthreads();
  for (int off = 128; off > 0; off >>= 1) {
    if (t < off) red[t] = fmaxf(red[t], red[t + off]);
    __syncthreads();
  }
  m = red[0];
  __syncthreads();

  float sum = 0.f;
#pragma unroll
  for (int u = 0; u < GC_N / 256; ++u)
    sum += __expf(sv[u] - m);
  red[t] = sum;
  __syncthreads();
  for (int off = 128; off > 0; off >>= 1) {
    if (t < off) red[t] += red[t + off];
    __syncthreads();
  }
  if (t == 0) {
    sm[i] = m;
    sinvl[i] = 1.0f / red[0];
  }
}

// ---------------------------------------------------------------------------
// Kernel 6 (fused): out = 0.5 * (softmax(S) @ H + H), A never materialized.
// Block = one 16-row stripe, 4 waves x 64 output columns each.
// Per k-step each lane rebuilds its 2-element A fragment from:
//   graph pair (b64), e pair (b64), c2 pair (b64)  + per-row scalars,
// applies exp only when an edge is present (pfill covers non-edges and the
// measure-zero empty-row case), then issues 4 f32 WMMAs.
// ---------------------------------------------------------------------------
__global__ void gc_fused_attn_gemm(const float* __restrict__ g,
                                   const float* __restrict__ e,
                                   const float* __restrict__ c1,
                                   const float* __restrict__ c2,
                                   const float* __restrict__ sm,
                                   const float* __restrict__ sinvl,
                                   const float* __restrict__ H,
                                   float* __restrict__ out) {
  const int tm = blockIdx.x;               // 16-row stripe, 0..255
  const int wv = threadIdx.x >> 5;         // 0..3 -> 64-column group
  const int lane = threadIdx.x & 31;
  const int l16 = lane & 15;
  const int khalf = (lane >> 4) << 1;      // 0 or 2

  const int m = (tm << 4) + l16;           // global row held by this lane (A frag)
  const int nbase = (wv << 6) + l16;       // column for tile j: nbase + 16*j

  const float ei    = e[m];
  const float ci    = c1[m];
  const float mi    = sm[m];
  const float invli = sinvl[m];
  // non-edge probability: exactly 0 unless the whole row was NEG_FILL
  const float pfill = (mi <= -1e10f) ? invli : 0.f;

  const float* __restrict__ grow = g + (size_t)m * GC_N + khalf;

  v8f acc[4] = {};
  for (int k = 0; k < GC_N; k += 4) {
    const int k0 = k + khalf;
    v2f g2 = *(const v2f*)(grow + k);      // graph[m][k0], graph[m][k0+1]
    v2f a;
    a.x = pfill;
    a.y = pfill;
    if (g2.x > 0.f && k0 != m) {
      v2f ee = *(const v2f*)(e + k0);
      v2f qq = *(const v2f*)(c2 + k0);
      float s = (ei * ee.x * g2.x) * (ci + qq.x) * 0.5f;
      a.x = __expf(s - mi) * invli;
    }
    if (g2.y > 0.f && (k0 + 1) != m) {
      v2f ee = *(const v2f*)(e + k0);
      v2f qq = *(const v2f*)(c2 + k0);
      float s = (ei * ee.y * g2.y) * (ci + qq.y) * 0.5f;
      a.y = __expf(s - mi) * invli;
    }
    const float* __restrict__ b0 = H + (size_t)k0 * GC_OUTF + nbase;
#pragma unroll
    for (int j = 0; j < 4; ++j) {
      v2f b;
      b.x = b0[j * 16];
      b.y = b0[GC_OUTF + j * 16];
      acc[j] = wmma_f32(a, b, acc[j]);
    }
  }

  const int row0 = (tm << 4) + ((lane >> 4) << 3);
#pragma unroll
  for (int j = 0; j < 4; ++j) {
#pragma unroll
    for (int r = 0; r < 8; ++r) {
      size_t idx = (size_t)(row0 + r) * GC_OUTF + nbase + j * 16;
      out[idx] = 0.5f * (acc[j][r] + H[idx]);
    }
  }
}

// ---------------------------------------------------------------------------
extern "C" void kernel_launch(void* const* d_in, const int* in_sizes, int n_in,
                              void* d_out, int out_size, void* d_ws, size_t ws_size,
                              hipStream_t stream) {
  const float* x      = (const float*)d_in[0];  // [N, INF]
  const float* graph  = (const float*)d_in[1];  // [N, N]
  const float* weight = (const float*)d_in[2];  // [OUTF, INF]
  const float* t_s    = (const float*)d_in[3];  // [OUTF]
  const float* t_r    = (const float*)d_in[4];  // [OUTF]
  float* out = (float*)d_out;                   // [N, OUTF]

  // Workspace layout (floats): H | c1 | c2 | d1 | e | sm | sinvl  (~4.3 MB)
  float* H     = (float*)d_ws;
  float* c1    = H    + (size_t)GC_N * GC_OUTF;
  float* c2    = c1   + GC_N;
  float* d1v   = c2   + GC_N;
  float* ev    = d1v  + GC_N;
  float* smv   = ev   + GC_N;
  float* sinvl = smv  + GC_N;

  // 1) H = x @ W^T   (4096 wave-tiles, 4 waves/block)
  gc_gemm_xwT<<<1024, 128, 0, stream>>>(x, weight, H);
  // 2) per-row attention scalars
  gc_rowdot<<<GC_N, 256, 0, stream>>>(H, t_s, t_r, c1, c2);
  // 3+4) double-normalization scales
  gc_deg1<<<GC_N, 256, 0, stream>>>(graph, d1v);
  gc_deg2<<<GC_N, 256, 0, stream>>>(graph, d1v, ev);
  // 5) softmax stats (row max + inverse partition)
  gc_attn_stats<<<GC_N, 256, 0, stream>>>(graph, ev, c1, c2, smv, sinvl);
  // 6) fused: out = 0.5*(softmax(S)@H + H)  (256 stripes x 4 waves)
  gc_fused_attn_gemm<<<256, 128, 0, stream>>>(graph, ev, c1, c2, smv, sinvl, H, out);
}